// FASTScore_78915729096901
// MI455X (gfx1250) — compile-verified
//
#include <hip/hip_runtime.h>

#define TILE_W   64
#define TILE_H   32
#define HALO     3
#define LDS_W    72                    // x-window [bx-4, bx+68) -> 18 x 16B chunks
#define IN_H     (TILE_H + 2 * HALO)   // 38 rows: [by-3, by+35)
#define NTHREADS 256
#define ROWS_PT  8                     // consecutive rows per thread
#define NCHUNKS  (18 * IN_H)           // 684 b128 transfers (fast path)
#define NSCALAR  (LDS_W * IN_H)        // 2736 b32 transfers (x-edge path)

// ---- CDNA5 feature probes -------------------------------------------------
#if defined(__has_builtin)
#  if __has_builtin(__builtin_amdgcn_global_load_async_to_lds_b32)
#    define HAVE_ASYNC_LDS 1
#  endif
#  if __has_builtin(__builtin_amdgcn_global_load_async_to_lds_b128)
#    define HAVE_ASYNC_LDS128 1
#  endif
#  if __has_builtin(__builtin_amdgcn_s_wait_asynccnt)
#    define HAVE_WAIT_ASYNC 1
#  endif
#endif

typedef int v4i __attribute__((ext_vector_type(4)));
typedef __attribute__((address_space(1))) int* gas_iptr;
typedef __attribute__((address_space(3))) int* lds_iptr;
typedef __attribute__((address_space(1))) v4i* gas_v4ptr;
typedef __attribute__((address_space(3))) v4i* lds_v4ptr;

// Run-of->=9 detector on a circular 16-bit mask. Returns nonzero iff a run
// exists. (Unmasked: bits 16..23 of r duplicate windows 0..7, so r!=0 is the
// exact predicate -- no final & 0xFFFF needed.)
__device__ __forceinline__ unsigned run9(unsigned b16) {
    unsigned x = b16 | (b16 << 16);  // circular duplication
    unsigned r = x & (x >> 1);       // runs >= 2
    r &= r >> 2;                     // runs >= 4
    r &= r >> 4;                     // runs >= 8
    r &= x >> 8;                     // runs >= 9
    return r;
}

__global__ __launch_bounds__(NTHREADS)
void fast_detect_kernel(const float* __restrict__ img, float* __restrict__ out,
                        int H, int W) {
    __shared__ __align__(16) float tile[IN_H * LDS_W];

    const int n  = blockIdx.z;
    const int bx = blockIdx.x * TILE_W;
    const int by = blockIdx.y * TILE_H;
    const int t  = threadIdx.x;

    const float* src = img + (size_t)n * H * W;

    // Only x-clamping breaks b128 contiguity; y replicate-pad is a row clamp.
    const bool xInterior = (bx >= 4) && (bx + 68 <= W);

#if defined(HAVE_ASYNC_LDS) && defined(HAVE_ASYNC_LDS128)
    if (xInterior) {
        // ---- Fast path: 16B-aligned b128 async copies, per-row y clamp ----
        const float* srcX = src + (bx - 4);
        auto load_chunk = [&](int q) {           // q in [0, 684)
            int ly  = q / 18;
            int lx4 = q - ly * 18;               // 16B chunk within row
            int gy  = by - 3 + ly;
            gy = gy < 0 ? 0 : (gy >= H ? H - 1 : gy);
            const float* p = srcX + (size_t)gy * W + lx4 * 4;
            __builtin_amdgcn_global_load_async_to_lds_b128(
                (gas_v4ptr)p, (lds_v4ptr)&tile[ly * LDS_W + lx4 * 4], 0, 0);
        };
#pragma unroll
        for (int i = 0; i < NCHUNKS / NTHREADS; ++i)   // 2 full rounds
            load_chunk(t + i * NTHREADS);
        if (t < NCHUNKS - (NCHUNKS / NTHREADS) * NTHREADS)  // 172-lane tail
            load_chunk(t + (NCHUNKS / NTHREADS) * NTHREADS);
    } else
#endif
    {
        // ---- x-edge path: per-element clamped loads (replicate padding) ----
        auto load_one = [&](int e) {             // e in [0, 2736)
            int ly = e / LDS_W;
            int lx = e - ly * LDS_W;
            int gx = bx - 4 + lx;
            int gy = by - 3 + ly;
            gx = gx < 0 ? 0 : (gx >= W ? W - 1 : gx);
            gy = gy < 0 ? 0 : (gy >= H ? H - 1 : gy);
            const float* p = src + (size_t)gy * W + gx;
#if defined(HAVE_ASYNC_LDS)
            __builtin_amdgcn_global_load_async_to_lds_b32(
                (gas_iptr)p, (lds_iptr)&tile[ly * LDS_W + lx], 0, 0);
#else
            tile[ly * LDS_W + lx] = *p;
#endif
        };
#pragma unroll
        for (int i = 0; i < NSCALAR / NTHREADS; ++i)   // 10 full rounds
            load_one(t + i * NTHREADS);
        if (t < NSCALAR - (NSCALAR / NTHREADS) * NTHREADS)  // 176-lane tail
            load_one(t + (NSCALAR / NTHREADS) * NTHREADS);
    }

#if defined(HAVE_ASYNC_LDS)
#  if defined(HAVE_WAIT_ASYNC)
    __builtin_amdgcn_s_wait_asynccnt(0);
#  else
    asm volatile("s_wait_asynccnt 0" ::: "memory");
#  endif
#endif
    __syncthreads();

    // ---- Compute: 8 consecutive rows per thread (64x32 tile / 256 threads)
    constexpr int DY[16] = { 0, 1, 2, 3, 3, 3, 2, 1, 0,-1,-2,-3,-3,-3,-2,-1};
    constexpr int DX[16] = {-3,-3,-2,-1, 0, 1, 2, 3, 3, 3, 2, 1, 0,-1,-2,-3};

    const int tx  = t & (TILE_W - 1);
    const int ty0 = t >> 6;                 // 0..3
    const int ox  = bx + tx;
    const bool xok = ox < W;

    const float thrP =  20.0f;
    const float thrN = -20.0f;

    // center column of this thread sits at LDS x-index tx+4 (window origin bx-4)
    const float* base = &tile[(ty0 * ROWS_PT + HALO) * LDS_W + (tx + 4)];

#pragma unroll
    for (int rr = 0; rr < ROWS_PT; ++rr) {
        const float* ctr = base + rr * LDS_W;
        const float c = *ctr;

        unsigned dark = 0u, bright = 0u;
#pragma unroll
        for (int i = 15; i >= 0; --i) {     // descending: mask = 2*mask + bit_i
            float v = ctr[DY[i] * LDS_W + DX[i]];
            float d = v - c;                // must match reference rounding
            unsigned sc0, sc1;
            // 2 ops/bit: cmp -> sgpr carry, then acc = 2*acc + carry
            asm("v_cmp_ge_f32_e64 %2, %4, %5\n\t"
                "v_add_co_ci_u32_e64 %0, %2, %0, %0, %2\n\t"
                "v_cmp_le_f32_e64 %3, %4, %6\n\t"
                "v_add_co_ci_u32_e64 %1, %3, %1, %1, %3"
                : "+v"(dark), "+v"(bright), "=&s"(sc0), "=&s"(sc1)
                : "v"(d), "v"(thrP), "v"(thrN));
        }

        unsigned hit = run9(dark) | run9(bright);
        const int oy = by + ty0 * ROWS_PT + rr;
        if (xok && oy < H)
            out[((size_t)n * H + oy) * W + ox] = hit ? 1.0f : 0.0f;
    }
}

extern "C" void kernel_launch(void* const* d_in, const int* in_sizes, int n_in,
                              void* d_out, int out_size, void* d_ws, size_t ws_size,
                              hipStream_t stream) {
    const float* img = (const float*)d_in[0];
    float* out = (float*)d_out;
    const int H = 1080, W = 1920;
    const int N = in_sizes[0] / (H * W);   // 4 for the reference shapes

    dim3 block(NTHREADS);
    dim3 grid((W + TILE_W - 1) / TILE_W, (H + TILE_H - 1) / TILE_H, N);
    hipLaunchKernelGGL(fast_detect_kernel, grid, block, 0, stream, img, out, H, W);
}